// NystromNonLinearKernel_34428457845093
// MI455X (gfx1250) — compile-verified
//
#include <hip/hip_runtime.h>
#include <math.h>

typedef __attribute__((ext_vector_type(2))) float v2f;
typedef __attribute__((ext_vector_type(8))) float v8f;

constexpr int kB    = 2;
constexpr int kCin  = 32;
constexpr int kH    = 64;
constexpr int kW    = 64;
constexpr int kJ    = kH * kW;   // 4096
constexpr int kHid  = 64;
constexpr int kCout = 32;
constexpr int kS    = 256;
constexpr int kTileJ   = 128;    // j-rows per block (8 waves x 16)
constexpr int kThreads = 256;

__device__ __forceinline__ float gelu_exact(float x) {
    // reference uses approximate=False -> erf-based gelu
    return 0.5f * x * (1.0f + erff(x * 0.70710678118654752440f));
}

// cw[j*64+h] = x_j*W1[0,h] + y_j*W1[1,h]
__global__ void prep_cw_kernel(const float* __restrict__ W1, float* __restrict__ cw) {
    int tid = blockIdx.x * blockDim.x + threadIdx.x;
    if (tid >= kJ * kHid) return;
    int j = tid >> 6, h = tid & 63;
    const float inv63 = 1.0f / 63.0f;
    float cx = (float)(j & 63) * inv63;   // x = (j % W)/63
    float cy = (float)(j >> 6) * inv63;   // y = (j / W)/63
    cw[tid] = cx * W1[h] + cy * W1[kHid + h];
}

// K[(b*S+s)*64+h] = b1[h] - x_s*W1[0,h] - y_s*W1[1,h] + sum_c v[b,c,idx_s]*W1[2+c,h]
__global__ void prep_k_kernel(const float* __restrict__ v, const int* __restrict__ indices,
                              const float* __restrict__ W1, const float* __restrict__ b1,
                              float* __restrict__ Kmat) {
    int tid = blockIdx.x * blockDim.x + threadIdx.x;
    if (tid >= kB * kS * kHid) return;
    int h  = tid & 63;
    int bs = tid >> 6;
    int b  = bs >> 8;        // S == 256
    int s  = bs & 255;
    int idx = indices[s];
    const float inv63 = 1.0f / 63.0f;
    float sx = (float)(idx & 63) * inv63;
    float sy = (float)(idx >> 6) * inv63;
    float acc = b1[h] - sx * W1[h] - sy * W1[kHid + h];
    #pragma unroll
    for (int c = 0; c < kCin; ++c)
        acc = fmaf(v[((b * kCin + c) << 12) + idx], W1[(2 + c) * kHid + h], acc);
    Kmat[tid] = acc;
}

// Main: G[b,j,h] = sum_s gelu(cw[j,h] + K[b,s,h]); out[b,c,j] = (G @ W2)[c]/S + b2[c]
__global__ void __launch_bounds__(kThreads)
nystrom_main_kernel(const float* __restrict__ cw, const float* __restrict__ Kmat,
                    const float* __restrict__ W2, const float* __restrict__ b2,
                    float* __restrict__ out) {
    extern __shared__ float smem[];
    float* Klds = smem;                        // S*HID = 16384 floats (64 KB)
    float* w2s  = smem + kS * kHid;            // HID*COUT = 2048 floats (8 KB)
    float* b2s  = w2s + kHid * kCout;          // 32 floats

    const int tid = threadIdx.x;
    const int b   = blockIdx.y;

    // Stage K[b], W2, b2 into LDS (vectorized b128)
    {
        const float4* Kg4  = (const float4*)(Kmat + (size_t)b * kS * kHid);
        float4* Kl4 = (float4*)Klds;
        for (int i = tid; i < (kS * kHid) / 4; i += kThreads) Kl4[i] = Kg4[i];
        const float4* Wg4 = (const float4*)W2;
        float4* Wl4 = (float4*)w2s;
        for (int i = tid; i < (kHid * kCout) / 4; i += kThreads) Wl4[i] = Wg4[i];
        if (tid < kCout) b2s[tid] = b2[tid];
    }
    __syncthreads();

    const int lane  = tid & 31;
    const int wave  = tid >> 5;
    const int jl    = lane & 15;            // A-fragment M index
    const int hoff  = (lane >> 4) << 1;     // 0 for lanes 0-15, 2 for lanes 16-31
    const int jbase = blockIdx.x * kTileJ + wave * 16;
    const int j     = jbase + jl;

    // Per-lane cw in registers, laid out to match WMMA A 16x4 f32 fragments:
    // fragment kk covers h = 4kk..4kk+3; this lane owns h = 4kk+hoff, 4kk+hoff+1.
    float2 cwreg[16];
    #pragma unroll
    for (int kk = 0; kk < 16; ++kk)
        cwreg[kk] = *(const float2*)(cw + (size_t)j * kHid + 4 * kk + hoff);

    float acc0[16], acc1[16];
    #pragma unroll
    for (int kk = 0; kk < 16; ++kk) { acc0[kk] = 0.0f; acc1[kk] = 0.0f; }

    // Dominant loop: 256 samples x 32 gelu per lane. LDS reads broadcast
    // (all lanes of a half-wave read the same address).
    const float2* K2 = (const float2*)Klds;
    const int lhalf = hoff >> 1;  // 0 or 1
    for (int s = 0; s < kS; ++s) {
        const float2* row = K2 + s * (kHid / 2) + lhalf;
        #pragma unroll
        for (int kk = 0; kk < 16; ++kk) {
            float2 kv = row[2 * kk];
            acc0[kk] += gelu_exact(cwreg[kk].x + kv.x);
            acc1[kk] += gelu_exact(cwreg[kk].y + kv.y);
        }
    }

    // G(16x64) @ W2(64x32) via 32x V_WMMA_F32_16X16X4_F32 (two 16-col tiles).
    v8f d0 = {}; v8f d1 = {};
    const int n = jl;
    #pragma unroll
    for (int kk = 0; kk < 16; ++kk) {
        v2f a; a.x = acc0[kk]; a.y = acc1[kk];
        int r = 4 * kk + hoff;   // B rows: lanes 0-15 -> K=4kk,4kk+1 ; lanes 16-31 -> K=4kk+2,4kk+3
        v2f bf0, bf1;
        bf0.x = w2s[r * kCout + n];            bf0.y = w2s[(r + 1) * kCout + n];
        bf1.x = w2s[r * kCout + 16 + n];       bf1.y = w2s[(r + 1) * kCout + 16 + n];
        d0 = __builtin_amdgcn_wmma_f32_16x16x4_f32(false, a, false, bf0, (short)0, d0, false, false);
        d1 = __builtin_amdgcn_wmma_f32_16x16x4_f32(false, a, false, bf1, (short)0, d1, false, false);
    }

    // D layout: VGPR r -> M = r (+8 for lanes 16-31), N = lane&15.
    // M maps to consecutive j -> two coalesced float4 stores per tile.
    const float scale = 1.0f / (float)kS;
    const int rowM = (lane >> 4) << 3;  // 0 or 8
    {
        float bc = b2s[n];
        float* o = out + ((size_t)(b * kCout + n)) * kJ + jbase + rowM;
        float4 lo = make_float4(d0[0] * scale + bc, d0[1] * scale + bc,
                                d0[2] * scale + bc, d0[3] * scale + bc);
        float4 hi = make_float4(d0[4] * scale + bc, d0[5] * scale + bc,
                                d0[6] * scale + bc, d0[7] * scale + bc);
        *(float4*)(o)     = lo;
        *(float4*)(o + 4) = hi;
    }
    {
        float bc = b2s[16 + n];
        float* o = out + ((size_t)(b * kCout + 16 + n)) * kJ + jbase + rowM;
        float4 lo = make_float4(d1[0] * scale + bc, d1[1] * scale + bc,
                                d1[2] * scale + bc, d1[3] * scale + bc);
        float4 hi = make_float4(d1[4] * scale + bc, d1[5] * scale + bc,
                                d1[6] * scale + bc, d1[7] * scale + bc);
        *(float4*)(o)     = lo;
        *(float4*)(o + 4) = hi;
    }
}

extern "C" void kernel_launch(void* const* d_in, const int* in_sizes, int n_in,
                              void* d_out, int out_size, void* d_ws, size_t ws_size,
                              hipStream_t stream) {
    const float* v       = (const float*)d_in[0];
    const int*   indices = (const int*)d_in[1];
    const float* W1      = (const float*)d_in[2];
    const float* b1      = (const float*)d_in[3];
    const float* W2      = (const float*)d_in[4];
    const float* b2      = (const float*)d_in[5];
    float* out = (float*)d_out;

    // Workspace: cW (J*HID floats = 1 MB) then K (B*S*HID floats = 128 KB)
    float* cw   = (float*)d_ws;
    float* Kmat = cw + (size_t)kJ * kHid;

    prep_cw_kernel<<<(kJ * kHid + 255) / 256, 256, 0, stream>>>(W1, cw);
    prep_k_kernel<<<(kB * kS * kHid + 255) / 256, 256, 0, stream>>>(v, indices, W1, b1, Kmat);

    size_t shbytes = (size_t)(kS * kHid + kHid * kCout + kCout) * sizeof(float); // ~72 KB, fine for 320 KB WGP LDS
    dim3 grid(kJ / kTileJ, kB);
    nystrom_main_kernel<<<grid, kThreads, shbytes, stream>>>(cw, Kmat, W2, b2, out);
}